// ddstgcn_15195594293350
// MI455X (gfx1250) — compile-verified
//
#include <hip/hip_runtime.h>

typedef __attribute__((ext_vector_type(2))) float v2f;
typedef __attribute__((ext_vector_type(8))) float v8f;

// Problem constants (from reference)
#define B_   32
#define C_   40
#define N_   512
#define L_   12
#define F_   480     // C_*L_ flattened columns
#define CO_  20
#define LO_  10

// GEMM tiling
#define MT   128
#define NT   16
#define KB   32
#define PAD  4       // LDS row stride = KB+PAD = 36 floats (conflict-free frag reads)

// ---------------------------------------------------------------------------
// Kernel 1: x[B][C][N][L] -> Xt[B][N][C*L]  (row-major GEMM operand)
// ---------------------------------------------------------------------------
__global__ __launch_bounds__(256) void k_transpose(const float* __restrict__ x,
                                                   float* __restrict__ xt) {
    const int w  = blockIdx.x;   // 0..511
    const int bb = blockIdx.y;   // 0..31
    const float* xb = x  + (size_t)bb * C_ * N_ * L_;
    float*       xo = xt + ((size_t)bb * N_ + w) * F_;
    for (int f = threadIdx.x; f < F_; f += 256) {
        const int c = f / L_;
        const int l = f - c * L_;
        xo[f] = xb[(size_t)c * N_ * L_ + (size_t)w * L_ + l];
    }
}

// ---------------------------------------------------------------------------
// Kernel 2: Y[b] = A[b](512x512) @ X[b](512x480), fp32 WMMA 16x16x4.
// Block: 256 thr = 8 waves; tile MTxNT = 128x16; wave i -> rows [16i,16i+16).
// ---------------------------------------------------------------------------
__global__ __launch_bounds__(256) void k_gemm_wmma(const float* __restrict__ A,
                                                   const float* __restrict__ X,
                                                   float* __restrict__ Y) {
    __shared__ float sA[MT][KB + PAD];      // A tile, row-major, padded
    __shared__ float sB[NT][KB + PAD];      // X tile, stored transposed [n][k]

    const int mt = blockIdx.x;              // 0..3
    const int nt = blockIdx.y;              // 0..29
    const int bb = blockIdx.z;              // 0..31

    const float* Ab = A + (size_t)bb * N_ * N_ + (size_t)mt * MT * N_;
    const float* Xb = X + (size_t)bb * N_ * F_ + nt * NT;
    float*       Yb = Y + (size_t)bb * N_ * F_ + (size_t)mt * MT * F_ + nt * NT;

    const int t    = threadIdx.x;
    const int wave = t >> 5;                // 0..7 (wave32)
    const int lane = t & 31;
    const int hi   = lane >> 4;             // lane-half selects K+2
    const int ln   = lane & 15;

    v8f acc = {};

    for (int kk = 0; kk < N_; kk += KB) {
        // Stage A tile: 128x32 floats = 1024 float4, 4 per thread, coalesced.
#pragma unroll
        for (int i = 0; i < 4; ++i) {
            const int idx = t + i * 256;        // 0..1023
            const int row = idx >> 3;
            const int c4  = (idx & 7) << 2;
            const float4 v = *(const float4*)(Ab + (size_t)row * N_ + kk + c4);
            *(float4*)&sA[row][c4] = v;         // (row*36 + c4)*4 is 16B aligned
        }
        // Stage B tile transposed: 32x16 floats, 2 per thread.
#pragma unroll
        for (int i = 0; i < 2; ++i) {
            const int idx = t + i * 256;        // 0..511
            const int kr  = idx >> 4;           // 0..31
            const int nc  = idx & 15;
            sB[nc][kr] = Xb[(size_t)(kk + kr) * F_ + nc];
        }
        __syncthreads();

        // 8 x V_WMMA_F32_16X16X4_F32 per K-block.
#pragma unroll
        for (int k4 = 0; k4 < KB; k4 += 4) {
            const v2f a = *(const v2f*)&sA[wave * 16 + ln][k4 + 2 * hi];
            const v2f b = *(const v2f*)&sB[ln][k4 + 2 * hi];
            acc = __builtin_amdgcn_wmma_f32_16x16x4_f32(
                /*neg_a=*/false, a, /*neg_b=*/false, b,
                /*c_mod=*/(short)0, acc, /*reuse_a=*/false, /*reuse_b=*/false);
        }
        __syncthreads();
    }

    // C/D layout: VGPR r -> rows r (lanes 0-15) and r+8 (lanes 16-31), col = ln.
#pragma unroll
    for (int r = 0; r < 8; ++r) {
        Yb[(size_t)(wave * 16 + r + 8 * hi) * F_ + ln] = acc[r];
    }
}

// ---------------------------------------------------------------------------
// Kernel 3: accumulate one channel-block's conv contribution into y.
// H: [B][N][C_*L_] block kb; W: [20][280][2]; y: [B][20][N][10].
// Block handles 16 consecutive v for one batch -> coalesced y access.
// ---------------------------------------------------------------------------
__global__ __launch_bounds__(256) void k_conv_accum(const float* __restrict__ H,
                                                    const float* __restrict__ W,
                                                    const float* __restrict__ bias,
                                                    float* __restrict__ y,
                                                    int kb, int init) {
    __shared__ float sH[16][F_];            // 16 rows of 480
    __shared__ float sW[CO_][C_][2];        // 20x40x2 weight slice

    const int v0 = blockIdx.x * 16;         // grid.x = 32
    const int bb = blockIdx.y;              // grid.y = 32
    const int t  = threadIdx.x;

    const float* Hr = H + ((size_t)bb * N_ + v0) * F_;
    for (int i = t; i < 16 * F_; i += 256) {
        sH[i / F_][i % F_] = Hr[i];
    }
    for (int i = t; i < CO_ * C_ * 2; i += 256) {
        const int o   = i / (C_ * 2);
        const int rem = i - o * (C_ * 2);
        const int c   = rem >> 1;
        const int tap = rem & 1;
        sW[o][c][tap] = W[(size_t)o * 280 * 2 + (size_t)(kb * C_ + c) * 2 + tap];
    }
    __syncthreads();

    // 20*16*10 = 3200 outputs; consecutive idx -> contiguous y within an o.
    for (int idx = t; idx < CO_ * 16 * LO_; idx += 256) {
        const int o    = idx / (16 * LO_);
        const int r    = idx - o * (16 * LO_);
        const int vloc = r / LO_;
        const int l    = r - vloc * LO_;
        const size_t yi = (((size_t)bb * CO_ + o) * N_ + (v0 + vloc)) * LO_ + l;
        float acc = init ? bias[o] : y[yi];
#pragma unroll
        for (int c = 0; c < C_; ++c) {
            acc += sH[vloc][c * L_ + l]     * sW[o][c][0];
            acc += sH[vloc][c * L_ + l + 2] * sW[o][c][1];
        }
        y[yi] = acc;
    }
}

// ---------------------------------------------------------------------------
// Launch: transpose; then per support s: Y1 = A_s@Xt (+conv), Y2 = A_s@Y1 (+conv)
// ---------------------------------------------------------------------------
extern "C" void kernel_launch(void* const* d_in, const int* in_sizes, int n_in,
                              void* d_out, int out_size, void* d_ws, size_t ws_size,
                              hipStream_t stream) {
    const float* x    = (const float*)d_in[0];
    const float* W    = (const float*)d_in[4];
    const float* bias = (const float*)d_in[5];
    float*       y    = (float*)d_out;

    const size_t nBuf = (size_t)B_ * N_ * F_;   // 7,864,320 floats each
    float* Xt = (float*)d_ws;
    float* Y1 = Xt + nBuf;
    float* Y2 = Y1 + nBuf;

    k_transpose<<<dim3(N_, B_), 256, 0, stream>>>(x, Xt);
    k_conv_accum<<<dim3(N_ / 16, B_), 256, 0, stream>>>(Xt, W, bias, y, 0, 1);

    for (int s = 0; s < 3; ++s) {
        const float* A = (const float*)d_in[1 + s];
        k_gemm_wmma<<<dim3(N_ / MT, F_ / NT, B_), 256, 0, stream>>>(A, Xt, Y1);
        k_conv_accum<<<dim3(N_ / 16, B_), 256, 0, stream>>>(Y1, W, bias, y, 1 + 2 * s, 0);
        k_gemm_wmma<<<dim3(N_ / MT, F_ / NT, B_), 256, 0, stream>>>(A, Y1, Y2);
        k_conv_accum<<<dim3(N_ / 16, B_), 256, 0, stream>>>(Y2, W, bias, y, 2 + 2 * s, 0);
    }
}